// WienerDeblurNet_82429012345583
// MI455X (gfx1250) — compile-verified
//
#include <hip/hip_runtime.h>
#include <math.h>

// ---------------- problem geometry ----------------
#define NP    532            // Hp = Wp = 512 + 2*PAD
#define NG    544            // GEMM-padded size = 34 * 16
#define NSUP  17             // 32x32 super-tiles per dim (NG/32)
#define NSUP2 (NSUP*NSUP)    // 289
#define N2G   (NG*NG)        // 295936
#define N2P   (NP*NP)        // 283024
#define B_    4
#define C_    3
#define NSL   (B_*C_)        // 12 (b,c) slices
#define KK    21
#define PADI  10
#define WF_   24
#define M_    3
#define IMG   512
#define IMG2  (IMG*IMG)

#define ANG(e) (-6.283185307179586f * (float)(e) / (float)NP)

typedef __attribute__((ext_vector_type(2))) float v2f;
typedef __attribute__((ext_vector_type(8))) float v8f;

// ---------------- small prep kernels ----------------

// normalized wiener weights: zero-mean, L2-normalized, * scale
__global__ void k_wnorm(const float* __restrict__ w, const float* __restrict__ sc,
                        float* __restrict__ wn) {
  int f = threadIdx.x;
  if (f >= WF_) return;
  float mean = 0.f;
  for (int j = 0; j < 25; ++j) mean += w[f*25+j];
  mean *= (1.0f/25.0f);
  float ss = 0.f;
  for (int j = 0; j < 25; ++j) { float t = w[f*25+j] - mean; ss += t*t; }
  float inv = sc[f] * rsqrtf(ss);
  for (int j = 0; j < 25; ++j) wn[f*25+j] = (w[f*25+j] - mean) * inv;
}

// DFT matrix F[u][k] = exp(-2*pi*i*u*k/532), zero-padded to 544x544.
// Phase computed via exact integer mod for precision.
__global__ void k_build_F(float* __restrict__ Fr, float* __restrict__ Fi) {
  int i = blockIdx.x * 256 + threadIdx.x;
  if (i >= N2G) return;
  int u = i / NG, k = i % NG;
  float cr = 0.f, ci = 0.f;
  if (u < NP && k < NP) {
    int e = (int)(((long long)u * (long long)k) % NP);
    float s, c; sincosf(ANG(e), &s, &c);
    cr = c; ci = s;
  }
  Fr[i] = cr; Fi[i] = ci;
}

// S(u,v) = sum_f |OTF_f(u,v)|^2 for the 24 normalized 5x5 filters (roll by 2)
__global__ void k_S(const float* __restrict__ wn, float* __restrict__ Sm) {
  int i = blockIdx.x * 256 + threadIdx.x;
  if (i >= N2P) return;
  int u = i / NP, v = i % NP;
  float pr[5], pi[5], qr[5], qi[5];
  for (int k = 0; k < 5; ++k) {
    int e = (int)(((long long)u * (long long)(k + NP - 2)) % NP);
    sincosf(ANG(e), &pi[k], &pr[k]);
    e = (int)(((long long)v * (long long)(k + NP - 2)) % NP);
    sincosf(ANG(e), &qi[k], &qr[k]);
  }
  float s = 0.f;
  for (int f = 0; f < WF_; ++f) {
    float gr = 0.f, gi = 0.f;
    for (int k = 0; k < 5; ++k)
      for (int l = 0; l < 5; ++l) {
        float wv = wn[f*25 + k*5 + l];
        gr += wv * (pr[k]*qr[l] - pi[k]*qi[l]);
        gi += wv * (pr[k]*qi[l] + pi[k]*qr[l]);
      }
    s += gr*gr + gi*gi;
  }
  Sm[u*NG + v] = s;
}

// Blur OTF per sample: Bf(u,v) = sum_{k,l} psf[k,l] exp(-2pi i (u(k-10)+v(l-10))/532)
__global__ void k_Bf(const float* __restrict__ bk, float* __restrict__ Bf) {
  int i = blockIdx.x * 256 + threadIdx.x;
  if (i >= N2P) return;
  int b = blockIdx.z;
  int u = i / NP, v = i % NP;
  const float* psf = bk + b * KK * KK;
  float cr[KK], ci[KK];
  for (int l = 0; l < KK; ++l) {
    int e = (int)(((long long)v * (long long)(l + NP - PADI)) % NP);
    sincosf(ANG(e), &ci[l], &cr[l]);
  }
  float ar = 0.f, ai = 0.f;
  for (int k = 0; k < KK; ++k) {
    int e = (int)(((long long)u * (long long)(k + NP - PADI)) % NP);
    float rs, rc; sincosf(ANG(e), &rs, &rc);
    float sr = 0.f, si = 0.f;
    for (int l = 0; l < KK; ++l) { float wv = psf[k*KK + l]; sr += wv*cr[l]; si += wv*ci[l]; }
    ar += rc*sr - rs*si;
    ai += rc*si + rs*sr;
  }
  size_t base = (size_t)b * 2 * N2G;
  Bf[base + (size_t)u*NG + v]       = ar;
  Bf[base + N2G + (size_t)u*NG + v] = ai;
}

// init per-sample min/max (nonneg floats -> unsigned compare is order-preserving)
__global__ void k_init_mm(unsigned* mm) {
  int t = threadIdx.x;
  if (t < 2*B_) mm[t] = (t & 1) ? 0u : 0x7F7FFFFFu;
}

__global__ void k_minmax(const float* __restrict__ img, unsigned* __restrict__ mm) {
  __shared__ unsigned smn, smx;
  int tid = threadIdx.x;
  size_t i = (size_t)blockIdx.x * 256 + tid;
  const size_t per = (size_t)C_ * IMG2;   // 786432, divisible by 256 -> block stays in one sample
  if (tid == 0) { smn = 0x7F7FFFFFu; smx = 0u; }
  __syncthreads();
  if (i < (size_t)B_ * per) {
    unsigned v = __float_as_uint(img[i]);
    atomicMin(&smn, v); atomicMax(&smx, v);
  }
  __syncthreads();
  if (tid == 0) {
    int b = (int)(((size_t)blockIdx.x * 256) / per);
    atomicMin(&mm[2*b], smn); atomicMax(&mm[2*b+1], smx);
  }
}

// symmetric pad 512 -> 532 per (b,c) slice
__global__ void k_pad(const float* __restrict__ img, float* __restrict__ xp) {
  long long i = (long long)blockIdx.x * 256 + threadIdx.x;
  if (i >= (long long)NSL * N2P) return;
  int s = (int)(i / N2P), r = (int)(i % N2P);
  int y = r / NP, x = r % NP;
  int sy = y - PADI; sy = (sy < 0) ? (-1 - sy) : ((sy >= IMG) ? (2*IMG - 1 - sy) : sy);
  int sx = x - PADI; sx = (sx < 0) ? (-1 - sx) : ((sx >= IMG) ? (2*IMG - 1 - sx) : sx);
  xp[i] = img[(size_t)s * IMG2 + (size_t)sy * IMG + sx];
}

// edgetaper betas: circular autocorrelation of psf projections over period 531
__global__ void k_beta(const float* __restrict__ bk, float* __restrict__ beta) {
  int b = blockIdx.x, axis = blockIdx.y;
  int t = threadIdx.x;
  if (t >= NP) return;
  const float* psf = bk + b * KK * KK;
  float p[KK];
  for (int j = 0; j < KK; ++j) {
    float s = 0.f;
    for (int o = 0; o < KK; ++o) s += (axis == 0) ? psf[j*KK + o] : psf[o*KK + j];
    p[j] = s;
  }
  float z0 = 0.f;
  for (int j = 0; j < KK; ++j) z0 += p[j]*p[j];
  const int n1 = NP - 1;                 // 531
  int tt = (t == n1) ? 0 : t;            // br[531] = z[0]
  float z = 0.f;
  if (tt == 0) z = z0;
  else if (tt <= KK-1) { for (int j = tt; j < KK; ++j) z += p[j]*p[j-tt]; }
  else if (tt >= n1-(KK-1)) { int d = n1 - tt; for (int j = 0; j <= KK-1-d; ++j) z += p[j]*p[j+d]; }
  beta[(b*2 + axis) * NG + t] = z / z0;
}

// edgetaper: out = clip(alpha*xp + (1-alpha)*circblur(xp,psf), min, max) -> XT (544-stride)
__global__ void k_taper(const float* __restrict__ xp, const float* __restrict__ bk,
                        const float* __restrict__ beta, const unsigned* __restrict__ mm,
                        float* __restrict__ XT) {
  long long i = (long long)blockIdx.x * 256 + threadIdx.x;
  if (i >= (long long)NSL * N2P) return;
  int s = (int)(i / N2P), r = (int)(i % N2P);
  int b = s / C_;
  int y = r / NP, x = r % NP;
  const float* xs  = xp + (size_t)s * N2P;
  const float* psf = bk + b * KK * KK;
  float blur = 0.f;
  for (int k = 0; k < KK; ++k) {
    int yy = y - (k - PADI); if (yy < 0) yy += NP; else if (yy >= NP) yy -= NP;
    const float* rowp = xs + (size_t)yy * NP;
    int xx = x + PADI; if (xx >= NP) xx -= NP;     // l = 0 -> x + 10
    for (int l = 0; l < KK; ++l) {
      blur += psf[k*KK + l] * rowp[xx];
      if (--xx < 0) xx += NP;
    }
  }
  float a = (1.f - beta[(b*2+0)*NG + y]) * (1.f - beta[(b*2+1)*NG + x]);
  float v = a * xs[r] + (1.f - a) * blur;
  float mn = __uint_as_float(mm[2*b]), mx = __uint_as_float(mm[2*b+1]);
  v = fminf(fmaxf(v, mn), mx);
  XT[(size_t)s * N2G + (size_t)y * NG + x] = v;
}

// ---------------- WMMA GEMM (DFT-as-matmul) ----------------
// Each wave computes a 32x32 super-tile (2x2 WMMA tiles, 4 f32 accumulators) of
// C = A0*B0 (+/-) A1*B1, all 544x544 row-major, K in steps of 4 via
// V_WMMA_F32_16X16X4_F32. Register tiling gives 2x reuse on both A and B
// fragments (~1.5 VMEM issues per WMMA), keeping the matrix pipe fed from L2.
// F32 WMMA has no A/B neg modifier (NEG = CNeg only), so the minus sign of the
// fused second product is applied by negating its A fragments with VALU.
#define WMMA4(a, b, c) __builtin_amdgcn_wmma_f32_16x16x4_f32(false, (a), false, (b), (short)0, (c), false, false)

template<bool HAS2, bool NEG2>
__device__ __forceinline__ void wave_gemm(const float* __restrict__ A0,
                                          const float* __restrict__ B0,
                                          const float* __restrict__ A1,
                                          const float* __restrict__ B1,
                                          float* __restrict__ C) {
  int wv   = threadIdx.x >> 5;
  int lane = threadIdx.x & 31;
  int st = blockIdx.x * 8 + wv;
  if (st >= NSUP2) return;                 // wave-uniform: EXEC stays all-ones
  int sm = st / NSUP, sn = st % NSUP;
  int half = lane >> 4, l16 = lane & 15;
  int r0 = sm*32 + l16;                    // A row, m-tile 0 (m-tile 1 = +16)
  int c0 = sn*32 + l16;                    // B col, n-tile 0 (n-tile 1 = +16)
  const float* a0m0 = A0 + (size_t)r0 * NG;
  const float* a0m1 = a0m0 + (size_t)16 * NG;
  const float* a1m0 = HAS2 ? (A1 + (size_t)r0 * NG) : nullptr;
  const float* a1m1 = HAS2 ? (a1m0 + (size_t)16 * NG) : nullptr;
  v8f acc00 = {}, acc01 = {}, acc10 = {}, acc11 = {};
  for (int k0 = 0; k0 < NG; k0 += 4) {
    int ka = k0 + half*2;                  // lanes 0-15: K=k0,k0+1 ; lanes 16-31: K=k0+2,k0+3
    {
      const float* br0 = B0 + (size_t)ka * NG;
      const float* br1 = B0 + (size_t)(ka+1) * NG;
      v2f am0, am1, bn0, bn1;
      am0.x = a0m0[ka]; am0.y = a0m0[ka+1];
      am1.x = a0m1[ka]; am1.y = a0m1[ka+1];
      bn0.x = br0[c0];      bn0.y = br1[c0];
      bn1.x = br0[c0 + 16]; bn1.y = br1[c0 + 16];
      acc00 = WMMA4(am0, bn0, acc00);
      acc01 = WMMA4(am0, bn1, acc01);
      acc10 = WMMA4(am1, bn0, acc10);
      acc11 = WMMA4(am1, bn1, acc11);
    }
    if (HAS2) {
      const float* br0 = B1 + (size_t)ka * NG;
      const float* br1 = B1 + (size_t)(ka+1) * NG;
      v2f am0, am1, bn0, bn1;
      am0.x = a1m0[ka]; am0.y = a1m0[ka+1];
      am1.x = a1m1[ka]; am1.y = a1m1[ka+1];
      if (NEG2) { am0.x = -am0.x; am0.y = -am0.y; am1.x = -am1.x; am1.y = -am1.y; }
      bn0.x = br0[c0];      bn0.y = br1[c0];
      bn1.x = br0[c0 + 16]; bn1.y = br1[c0 + 16];
      acc00 = WMMA4(am0, bn0, acc00);
      acc01 = WMMA4(am0, bn1, acc01);
      acc10 = WMMA4(am1, bn0, acc10);
      acc11 = WMMA4(am1, bn1, acc11);
    }
  }
  // store 4 tiles; C layout: VGPR r -> rows {r, r+8} for lane halves {0,1}
  {
    float* cp = C + (size_t)(sm*32 + half*8) * NG + (sn*32 + l16);
    #pragma unroll
    for (int r = 0; r < 8; ++r) cp[(size_t)r * NG] = acc00[r];
    cp += 16;
    #pragma unroll
    for (int r = 0; r < 8; ++r) cp[(size_t)r * NG] = acc01[r];
    cp = C + (size_t)(sm*32 + 16 + half*8) * NG + (sn*32 + l16);
    #pragma unroll
    for (int r = 0; r < 8; ++r) cp[(size_t)r * NG] = acc10[r];
    cp += 16;
    #pragma unroll
    for (int r = 0; r < 8; ++r) cp[(size_t)r * NG] = acc11[r];
  }
}

// forward pass 1: T[z] = XT[z/2] * F[z&1]  (real input x complex DFT matrix), z in [0,24)
__global__ void __launch_bounds__(256) k_gemm1(const float* __restrict__ XT,
                                               const float* __restrict__ F,
                                               float* __restrict__ T) {
  int z = blockIdx.z;
  wave_gemm<false, false>(XT + (size_t)(z >> 1) * N2G, F + (size_t)(z & 1) * N2G,
                          nullptr, nullptr, T + (size_t)z * N2G);
}

// generic fused 2-product GEMM: C[z] = A0[z]*B0[z] (+/-) A1[z]*B1[z]
template<bool NEG2>
__global__ void __launch_bounds__(256) k_gemm2(const float* __restrict__ A0, long long sA,
                                               const float* __restrict__ B0, long long sB,
                                               const float* __restrict__ A1,
                                               const float* __restrict__ B1,
                                               float* __restrict__ C, long long sC) {
  int z = blockIdx.z;
  wave_gemm<true, NEG2>(A0 + (size_t)z * sA, B0 + (size_t)z * sB,
                        A1 + (size_t)z * sA, B1 + (size_t)z * sB,
                        C + (size_t)z * sC);
}

// ---------------- pointwise Wiener + crop ----------------
__global__ void k_wiener(const float* __restrict__ Y, const float* __restrict__ Bf,
                         const float* __restrict__ Sm, const float* __restrict__ alpha,
                         int m, float* __restrict__ Xf) {
  int i = blockIdx.x * 256 + threadIdx.x;
  if (i >= N2G) return;
  int s = blockIdx.z, b = s / C_;
  int u = i / NG, v = i % NG;
  float xr = 0.f, xi = 0.f;
  if (u < NP && v < NP) {
    size_t yb = (size_t)s * 2 * N2G;
    size_t bb = (size_t)b * 2 * N2G;
    float yr = Y[yb + i],  yi = Y[yb + N2G + i];
    float br = Bf[bb + i], bi = Bf[bb + N2G + i];
    float a  = expf(alpha[m]);
    float inv = 1.0f / (br*br + bi*bi + a * Sm[i]);
    xr = (br*yr + bi*yi) * inv;           // conj(Bf)*Y / D
    xi = (br*yi - bi*yr) * inv;
  }
  Xf[(size_t)s * 2 * N2G + i]       = xr;
  Xf[(size_t)s * 2 * N2G + N2G + i] = xi;
}

__global__ void k_crop(const float* __restrict__ R, float* __restrict__ out) {
  long long i = (long long)blockIdx.x * 256 + threadIdx.x;
  if (i >= (long long)NSL * IMG2) return;
  int s = (int)(i / IMG2), r = (int)(i % IMG2);
  int y = r / IMG, x = r % IMG;
  out[i] += R[(size_t)s * N2G + (size_t)(y + PADI) * NG + (x + PADI)]
            * (1.0f / ((float)NP * (float)NP));   // ifft2 normalization
}

// ---------------- launch ----------------
extern "C" void kernel_launch(void* const* d_in, const int* in_sizes, int n_in,
                              void* d_out, int out_size, void* d_ws, size_t ws_size,
                              hipStream_t stream) {
  const float* img = (const float*)d_in[0];   // (4,3,512,512)
  const float* bk  = (const float*)d_in[1];   // (4,1,21,21)
  const float* wcv = (const float*)d_in[3];   // (24,1,5,5)
  const float* wsc = (const float*)d_in[4];   // (24,)
  const float* alp = (const float*)d_in[5];   // (3,1)
  float* out = (float*)d_out;
  float* W = (float*)d_ws;

  // workspace layout (floats); total ~126 MB
  float* Fr   = W;                               // N2G
  float* Fi   = W + (size_t)N2G;                 // N2G (contiguous after Fr: k_gemm1 relies on it)
  float* Sm   = W + 2*(size_t)N2G;               // N2G
  float* Bf   = W + 3*(size_t)N2G;               // B*2*N2G
  float* wn   = W + (3 + 2*(size_t)B_)*N2G;      // 600 (pad 1024)
  float* beta = wn + 1024;                       // B*2*NG
  unsigned* mm = (unsigned*)(beta + (size_t)B_*2*NG); // 8 (pad 16)
  float* xp   = (float*)(mm + 16);               // NSL*N2P
  float* XT   = xp + (size_t)NSL*N2P;            // NSL*N2G  (also reused as real ifft output)
  float* T    = XT + (size_t)NSL*N2G;            // NSL*2*N2G (fwd rows / inv rows)
  float* Yb   = T  + (size_t)NSL*2*N2G;          // NSL*2*N2G (forward FFT result)
  float* Xf   = Yb + (size_t)NSL*2*N2G;          // NSL*2*N2G (per-m filtered spectrum)

  const long long S0 = 0, S1 = (long long)N2G, S2 = 2LL*N2G;

  (void)hipMemsetAsync(out, 0, (size_t)out_size * sizeof(float), stream);
  (void)hipMemsetAsync(XT, 0, (size_t)NSL * N2G * sizeof(float), stream);

  k_init_mm<<<1, 32, 0, stream>>>(mm);
  k_wnorm  <<<1, 32, 0, stream>>>(wcv, wsc, wn);
  k_build_F<<<(N2G + 255)/256, 256, 0, stream>>>(Fr, Fi);
  k_S      <<<(N2P + 255)/256, 256, 0, stream>>>(wn, Sm);
  k_Bf     <<<dim3((N2P + 255)/256, 1, B_), 256, 0, stream>>>(bk, Bf);
  k_minmax <<<(B_*C_*IMG2 + 255)/256, 256, 0, stream>>>(img, mm);
  k_pad    <<<(NSL*N2P + 255)/256, 256, 0, stream>>>(img, xp);
  k_beta   <<<dim3(B_, 2), NG, 0, stream>>>(bk, beta);
  k_taper  <<<(NSL*N2P + 255)/256, 256, 0, stream>>>(xp, bk, beta, mm, XT);

  const int GB = (NSUP2 + 7) / 8;   // 37 blocks of 8 waves (one 32x32 super-tile each)

  // forward fft2: T = XT * F (real x complex), then Y = F * T (complex x complex)
  k_gemm1<<<dim3(GB, 1, 2*NSL), 256, 0, stream>>>(XT, Fr, T);
  k_gemm2<true ><<<dim3(GB, 1, NSL), 256, 0, stream>>>(Fr, S0, T,      S2, Fi, T + N2G, Yb,       S2); // Yr = Fr*Tr - Fi*Ti
  k_gemm2<false><<<dim3(GB, 1, NSL), 256, 0, stream>>>(Fr, S0, T + N2G,S2, Fi, T,       Yb + N2G, S2); // Yi = Fr*Ti + Fi*Tr

  for (int m = 0; m < M_; ++m) {
    k_wiener<<<dim3((N2G + 255)/256, 1, NSL), 256, 0, stream>>>(Yb, Bf, Sm, alp, m, Xf);
    // inverse pass 1: T2 = Xf * conj(F)
    k_gemm2<false><<<dim3(GB, 1, NSL), 256, 0, stream>>>(Xf,       S2, Fr, S0, Xf + N2G, Fi, T,       S2); // T2r = Xfr*Fr + Xfi*Fi
    k_gemm2<true ><<<dim3(GB, 1, NSL), 256, 0, stream>>>(Xf + N2G, S2, Fr, S0, Xf,       Fi, T + N2G, S2); // T2i = Xfi*Fr - Xfr*Fi
    // inverse pass 2, real part only: R = Fr*T2r + Fi*T2i  (into XT, reused)
    k_gemm2<false><<<dim3(GB, 1, NSL), 256, 0, stream>>>(Fr, S0, T, S2, Fi, T + N2G, XT, S1);
    k_crop<<<(NSL*IMG2 + 255)/256, 256, 0, stream>>>(XT, out);
  }
}